// ResidualAttentionBlock_58523224375613
// MI455X (gfx1250) — compile-verified
//
#include <hip/hip_runtime.h>
#include <hip/hip_bf16.h>

typedef __attribute__((ext_vector_type(2))) float v2f;
typedef __attribute__((ext_vector_type(8))) float v8f;

// D = A(16x4) * B(4x16) + C, full f32 precision on the matrix pipe.
__device__ __forceinline__ v8f wmma_f32_4(v2f a, v2f b, v8f c) {
  return __builtin_amdgcn_wmma_f32_16x16x4_f32(
      /*neg_a=*/false, a, /*neg_b=*/false, b,
      /*c_mod=*/(short)0, c, /*reuse_a=*/false, /*reuse_b=*/false);
}

#define BS 4
#define NTOK 256
#define DM 64
#define NH 4
#define M_TOT (BS * NTOK)     // 1024
#define N_QV 512              // q (256) + vf (256); vb unused by reference

// ---------------------------------------------------------------------------
// Kernel 1: v = x @ wqv_w.T + wqv_b  (only first 512 cols), fused q *= wk.
// One wave per 16x16 tile; K=64 -> 16 x v_wmma_f32_16x16x4_f32.
// ---------------------------------------------------------------------------
__global__ __launch_bounds__(256) void qv_proj_kernel(
    const float* __restrict__ x,       // [1024][64]
    const float* __restrict__ wk,      // [4][64] (flat 256)
    const float* __restrict__ wqv_w,   // [768][64]
    const float* __restrict__ wqv_b,   // [768]
    float* __restrict__ qs,            // [1024][256]  (q * wk)
    float* __restrict__ vf)            // [1024][256]
{
  const int lane  = threadIdx.x & 31;
  const int wave  = threadIdx.x >> 5;
  const int tile  = blockIdx.x * 8 + wave;        // 2048 tiles total
  const int m0    = (tile >> 5) << 4;             // M tile base (0..1008)
  const int n0    = (tile & 31) << 4;             // N tile base (0..496)
  const int mrow  = m0 + (lane & 15);             // A row for this lane
  const int ncol  = n0 + (lane & 15);             // B row (= output col)
  const int khalf = (lane >> 4) << 1;             // 0 or 2

  v8f acc = {0.f, 0.f, 0.f, 0.f, 0.f, 0.f, 0.f, 0.f};
  const float* xr = x + mrow * DM;
  const float* wr = wqv_w + ncol * DM;
  #pragma unroll
  for (int ks = 0; ks < DM; ks += 4) {
    v2f a, b;
    a.x = xr[ks + khalf];
    a.y = xr[ks + khalf + 1];
    b.x = wr[ks + khalf];
    b.y = wr[ks + khalf + 1];
    acc = wmma_f32_4(a, b, acc);
  }

  const float bias = wqv_b[ncol];
  const float scl  = (ncol < 256) ? wk[ncol] : 1.0f;  // wk[h*64+d] == wk[n]
  #pragma unroll
  for (int r = 0; r < 8; ++r) {
    const int m   = m0 + r + ((lane >> 4) << 3);
    const float v = acc[r] + bias;
    if (ncol < 256) qs[m * 256 + ncol] = v * scl;
    else            vf[m * 256 + (ncol - 256)] = v;
  }
}

// ---------------------------------------------------------------------------
// Kernel 2: fused L1-distance attention + softmax + AV (WMMA) + head-sum +
// SiLU + fan-in GEMM (WMMA, C preloaded with residual + bias).
// Grid: bs * (ntok/16) = 64 blocks of 256 threads (8 waves).
// ---------------------------------------------------------------------------
__global__ __launch_bounds__(256) void attn_fused_kernel(
    const float* __restrict__ x,        // [4][256][64]
    const float* __restrict__ wk,       // [4][64]
    const float* __restrict__ qs,       // [1024][256]
    const float* __restrict__ vf,       // [1024][256]
    const float* __restrict__ fanin_w,  // [64][64]
    const float* __restrict__ fanin_b,  // [64]
    float* __restrict__ out)            // [4][256][64]
{
  __shared__ float sh_q[NH * 16 * DM];   // 16 KB: q tile, all heads
  __shared__ float sh_att[NTOK * 16];    // 16 KB: scores/probs [i][j]
  __shared__ float sh_red[2 * 16];       // softmax max & 1/sum per column
  __shared__ float sh_b[16 * DM];        //  4 KB: head-summed context

  const int tid  = threadIdx.x;
  const int lane = tid & 31;
  const int wave = tid >> 5;
  const int b    = blockIdx.x >> 4;
  const int j0   = (blockIdx.x & 15) << 4;

  // Stage q_scaled tile (4 heads x 16 queries x 64) into LDS.
  for (int idx = tid; idx < NH * 16 * DM; idx += 256) {
    const int h = idx >> 10;
    const int j = (idx >> 6) & 15;
    const int d = idx & 63;
    sh_q[idx] = qs[(b * NTOK + j0 + j) * 256 + h * DM + d];
  }
  __syncthreads();

  v8f bacc = {0.f, 0.f, 0.f, 0.f, 0.f, 0.f, 0.f, 0.f};  // waves 0-3: AV acc

  const int i = tid;  // each thread owns one key token
  for (int h = 0; h < NH; ++h) {
    // ---- L1-distance scores: dist[i][j] = sum_d |q[j,d] - x[i,d]*wk[d]| ----
    float acc[16];
    #pragma unroll
    for (int j = 0; j < 16; ++j) acc[j] = 0.f;
    const float* xi  = x + (b * NTOK + i) * DM;
    const float* wkh = wk + h * DM;
    const float* qh  = sh_q + h * 16 * DM;
    for (int d = 0; d < DM; ++d) {
      const float kv = xi[d] * wkh[d];      // k = x * wk, on the fly
      #pragma unroll
      for (int j = 0; j < 16; ++j)
        acc[j] += fabsf(qh[j * DM + d] - kv);
    }
    #pragma unroll
    for (int j = 0; j < 16; ++j) {
      const float dist = acc[j] * 0.125f;   // / sqrt(dm)
      const float hub  = (dist < 0.2f) ? (0.5f * dist * dist)
                                       : (0.2f * (dist - 0.1f));
      sh_att[i * 16 + j] = -hub;
    }
    __syncthreads();

    // ---- softmax over key axis (per query column j) ----
    if (tid < 16) {
      const int j = tid;
      float mx = -3.0e38f;
      for (int i2 = 0; i2 < NTOK; ++i2)
        mx = fmaxf(mx, sh_att[i2 * 16 + j]);
      float s = 0.f;
      for (int i2 = 0; i2 < NTOK; ++i2)
        s += __expf(sh_att[i2 * 16 + j] - mx);
      sh_red[j]      = mx;
      sh_red[16 + j] = 1.0f / s;
    }
    __syncthreads();
    #pragma unroll
    for (int j = 0; j < 16; ++j)
      sh_att[i * 16 + j] =
          __expf(sh_att[i * 16 + j] - sh_red[j]) * sh_red[16 + j];
    __syncthreads();

    // ---- AV: waves 0-3 each do one 16(j) x 16(w) tile, K=256, f32 WMMA ----
    if (wave < 4) {
      const int w0    = (wave & 3) << 4;
      const int jm    = lane & 15;
      const int wn    = w0 + (lane & 15);
      const int khalf = (lane >> 4) << 1;
      const float* vfh = vf + (b * NTOK) * 256 + h * DM + wn;
      for (int ks = 0; ks < NTOK; ks += 4) {
        v2f a, bb;
        a.x  = sh_att[(ks + khalf) * 16 + jm];
        a.y  = sh_att[(ks + khalf + 1) * 16 + jm];
        bb.x = vfh[(ks + khalf) * 256];
        bb.y = vfh[(ks + khalf + 1) * 256];
        bacc = wmma_f32_4(a, bb, bacc);   // accumulates across heads too
      }
    }
    __syncthreads();  // protect sh_att before next head overwrites it
  }

  // ---- spill head-summed context to LDS ----
  if (wave < 4) {
    const int w0 = (wave & 3) << 4;
    #pragma unroll
    for (int r = 0; r < 8; ++r) {
      const int j = r + ((lane >> 4) << 3);
      sh_b[j * DM + w0 + (lane & 15)] = bacc[r];
    }
  }
  __syncthreads();

  // ---- SiLU (x * sigmoid(1.702 x)) in place ----
  for (int idx = tid; idx < 16 * DM; idx += 256) {
    const float v = sh_b[idx];
    sh_b[idx] = v / (1.0f + __expf(-1.702f * v));
  }
  __syncthreads();

  // ---- fan-in GEMM: out = x + fanin_b + silu(b) @ fanin_w.T ----
  if (wave < 4) {
    const int n0    = (wave & 3) << 4;
    const int ncol  = n0 + (lane & 15);
    const int jm    = lane & 15;
    const int khalf = (lane >> 4) << 1;

    v8f yacc;
    #pragma unroll
    for (int r = 0; r < 8; ++r) {        // C preloaded with residual + bias
      const int j = r + ((lane >> 4) << 3);
      yacc[r] = x[(b * NTOK + j0 + j) * DM + ncol] + fanin_b[ncol];
    }
    const float* fw = fanin_w + ncol * DM;
    #pragma unroll
    for (int ks = 0; ks < DM; ks += 4) {
      v2f a, bb;
      a.x  = sh_b[jm * DM + ks + khalf];
      a.y  = sh_b[jm * DM + ks + khalf + 1];
      bb.x = fw[ks + khalf];
      bb.y = fw[ks + khalf + 1];
      yacc = wmma_f32_4(a, bb, yacc);
    }
    #pragma unroll
    for (int r = 0; r < 8; ++r) {
      const int j = r + ((lane >> 4) << 3);
      out[(b * NTOK + j0 + j) * DM + ncol] = yacc[r];
    }
  }
}

extern "C" void kernel_launch(void* const* d_in, const int* in_sizes, int n_in,
                              void* d_out, int out_size, void* d_ws, size_t ws_size,
                              hipStream_t stream) {
  const float* x        = (const float*)d_in[0];
  const float* wk       = (const float*)d_in[1];
  const float* wqv_w    = (const float*)d_in[2];
  const float* wqv_b    = (const float*)d_in[3];
  const float* fanin_w  = (const float*)d_in[4];
  const float* fanin_b  = (const float*)d_in[5];
  float* out = (float*)d_out;

  float* qs = (float*)d_ws;            // 1024*256 floats = 1 MB
  float* vf = qs + M_TOT * 256;        // 1024*256 floats = 1 MB

  // Kernel 1: 2048 tiles / 8 waves per block = 256 blocks.
  qv_proj_kernel<<<256, 256, 0, stream>>>(x, wk, wqv_w, wqv_b, qs, vf);
  // Kernel 2: bs * ntok/16 = 64 blocks.
  attn_fused_kernel<<<64, 256, 0, stream>>>(x, wk, qs, vf, fanin_w, fanin_b, out);
}